// SpatialRelationshipEncoder_48679159333098
// MI455X (gfx1250) — compile-verified
//
#include <hip/hip_runtime.h>
#include <hip/hip_bf16.h>
#include <math.h>

// ---------------------------------------------------------------------------
// SpatialRelationshipEncoder for MI455X (gfx1250, wave32)
//
// Work breakdown (B=8, N=40000, D=256, S=128):
//   K0: zero per-(batch,superpoint) accumulators
//   K1: segment-sum of coords + counts (LDS-accumulated scatter-add)
//   K2: centers -> pairwise 128x128 distances -> rel[S,4], row mask
//   K3: MLP on rel using V_WMMA_F32_16X16X4_F32 (one workgroup / batch)
//   K4: out[b,n,:] = table[b, labels[b,n], :]  (327 MB stream, NT stores)
// ---------------------------------------------------------------------------

typedef __attribute__((ext_vector_type(2))) float v2f;
typedef __attribute__((ext_vector_type(4))) float f4;
typedef __attribute__((ext_vector_type(8))) float v8f;

static constexpr int B   = 8;
static constexpr int N   = 40000;
static constexpr int D   = 256;
static constexpr int HID = 64;
static constexpr int S   = 128;
static constexpr float LN_EPS = 1e-5f;

// workspace layout (float offsets)
static constexpr long WS_ACC = 0;                       // B*S*4  (sx,sy,sz,count)
static constexpr long WS_REL = WS_ACC + (long)B*S*4;    // B*S*4
static constexpr long WS_MSK = WS_REL + (long)B*S*4;    // B*S
static constexpr long WS_T1  = WS_MSK + (long)B*S;      // B*S*HID
static constexpr long WS_H   = WS_T1  + (long)B*S*HID;  // B*S*D
static constexpr long WS_A1  = WS_H   + (long)B*S*D;    // B*S*D
static constexpr long WS_G   = WS_A1  + (long)B*S*D;    // B*S*D
static constexpr long WS_TAB = WS_G   + (long)B*S*D;    // B*S*D  (final table)

__device__ __forceinline__ float gelu_exact(float x) {
    return x * 0.5f * (1.0f + erff(x * 0.70710678118654752440f));
}

// ----------------------------- K0: zero acc --------------------------------
__global__ void k0_zero_acc(float* __restrict__ ws) {
    int i = blockIdx.x * blockDim.x + threadIdx.x;
    if (i < B * S * 4) ws[WS_ACC + i] = 0.0f;
}

// ------------------------- K1: segment reduction ---------------------------
// grid = B * CHUNKS blocks, 256 threads. LDS accumulate, then global atomics.
static constexpr int CHUNKS = 64;

__global__ __launch_bounds__(256) void k1_segment_sum(
    const float* __restrict__ coords,      // [B,N,3]
    const long long* __restrict__ labels,  // [B,N] int64
    float* __restrict__ ws) {
    __shared__ float acc[S * 4];
    int b = blockIdx.x / CHUNKS;
    int chunk = blockIdx.x % CHUNKS;
    for (int i = threadIdx.x; i < S * 4; i += blockDim.x) acc[i] = 0.0f;
    __syncthreads();

    int per = (N + CHUNKS - 1) / CHUNKS;
    int n0 = chunk * per;
    int n1 = min(N, n0 + per);
    for (int n = n0 + threadIdx.x; n < n1; n += blockDim.x) {
        long base = ((long)b * N + n);
        int lab = (int)labels[base];
        float x = coords[base * 3 + 0];
        float y = coords[base * 3 + 1];
        float z = coords[base * 3 + 2];
        atomicAdd(&acc[lab * 4 + 0], x);
        atomicAdd(&acc[lab * 4 + 1], y);
        atomicAdd(&acc[lab * 4 + 2], z);
        atomicAdd(&acc[lab * 4 + 3], 1.0f);
    }
    __syncthreads();
    float* gacc = ws + WS_ACC + (long)b * S * 4;
    for (int i = threadIdx.x; i < S * 4; i += blockDim.x)
        if (acc[i] != 0.0f) atomicAdd(&gacc[i], acc[i]);
}

// ------------------ K2: centers, pairwise stats, rel -----------------------
// one block per batch, 128 threads (one per superpoint)
__global__ __launch_bounds__(S) void k2_rel(float* __restrict__ ws) {
    __shared__ float cx[S], cy[S], cz[S], vm[S];
    int b = blockIdx.x;
    int s = threadIdx.x;
    const float* gacc = ws + WS_ACC + (long)b * S * 4;

    float cnt = gacc[s * 4 + 3];
    float inv = 1.0f / fmaxf(cnt, 1.0f);
    cx[s] = gacc[s * 4 + 0] * inv;
    cy[s] = gacc[s * 4 + 1] * inv;
    cz[s] = gacc[s * 4 + 2] * inv;
    vm[s] = (cnt >= 2.0f) ? 1.0f : 0.0f;
    __syncthreads();

    float nv = 0.0f;
    for (int j = 0; j < S; j++) nv += vm[j];
    float denom = fmaxf(nv, 1.0f);

    float x = cx[s], y = cy[s], z = cz[s];
    float mean_d = 0.0f, frac = 0.0f, min_d = INFINITY;
    for (int j = 0; j < S; j++) {
        float dx = x - cx[j], dy = y - cy[j], dz = z - cz[j];
        float d2 = dx * dx + dy * dy + dz * dz;
        float dist = (d2 > 0.0f) ? sqrtf(d2) : 0.0f;
        mean_d += dist * vm[j];
        if (vm[j] > 0.0f) min_d = fminf(min_d, dist);
        frac += ((z > cz[j]) ? 1.0f : 0.0f) * vm[j];
    }
    mean_d /= denom;
    frac /= denom;

    float* rel = ws + WS_REL + ((long)b * S + s) * 4;
    rel[0] = mean_d;
    rel[1] = min_d;
    rel[2] = z;
    rel[3] = frac;
    float enough = (nv >= 2.0f) ? 1.0f : 0.0f;
    ws[WS_MSK + (long)b * S + s] = vm[s] * enough;
}

// --------------------------- K3: WMMA MLP ----------------------------------
// One workgroup per batch, 8 waves. GEMM M=128 fixed; tiles split over waves.
// V_WMMA_F32_16X16X4_F32 operand layouts (wave32, half = lane/16, r = lane%16):
//   A 16x4 : a[j]  = A[r][2*half + j]          (2 VGPRs)
//   B 4x16 : b[j]  = B[2*half + j][r]          (2 VGPRs)
//   C 16x16: c[i]  = C[i + 8*half][r]          (8 VGPRs)
__device__ __forceinline__ void gemm128_wmma(
    const float* __restrict__ X,   // [128,K] row-major
    int K,
    const float* __restrict__ W,   // [K,Nc] row-major
    const float* __restrict__ bias,// [Nc]
    float* __restrict__ Y,         // [128,Nc]
    int Nc,
    int act,                       // 1 = exact gelu
    const float* __restrict__ rowmask) { // nullable, [128]
    const int wave = threadIdx.x >> 5;
    const int lane = threadIdx.x & 31;
    const int hf   = lane >> 4;
    const int r    = lane & 15;
    const int ntiles = (Nc >> 4) * 8;

    for (int t = wave; t < ntiles; t += 8) {
        const int mi = t & 7;
        const int ni = t >> 3;
        const int row = mi * 16 + r;
        const int col = ni * 16 + r;
        v8f acc = {};
        for (int k = 0; k < K; k += 4) {
            const int ka = k + 2 * hf;
            v2f a = { X[row * K + ka], X[row * K + ka + 1] };
            v2f bm = { W[ka * Nc + col], W[(ka + 1) * Nc + col] };
            acc = __builtin_amdgcn_wmma_f32_16x16x4_f32(
                false, a, false, bm, (short)0, acc, false, false);
        }
        const float bv = bias[col];
#pragma unroll
        for (int i = 0; i < 8; i++) {
            const int orow = mi * 16 + i + 8 * hf;
            float v = acc[i] + bv;
            if (act == 1) v = gelu_exact(v);
            if (rowmask) v *= rowmask[orow];
            Y[(long)orow * Nc + col] = v;
        }
    }
}

__global__ __launch_bounds__(256) void k3_mlp(
    float* __restrict__ ws,
    const float* __restrict__ W1, const float* __restrict__ b1,
    const float* __restrict__ W2, const float* __restrict__ b2,
    const float* __restrict__ W3, const float* __restrict__ b3,
    const float* __restrict__ ln_g, const float* __restrict__ ln_b,
    const float* __restrict__ W4, const float* __restrict__ b4) {
    const int b = blockIdx.x;
    const float* relb = ws + WS_REL + (long)b * S * 4;
    const float* mskb = ws + WS_MSK + (long)b * S;
    float* t1b = ws + WS_T1 + (long)b * S * HID;
    float* hb  = ws + WS_H  + (long)b * S * D;
    float* a1b = ws + WS_A1 + (long)b * S * D;
    float* gb  = ws + WS_G  + (long)b * S * D;
    float* tab = ws + WS_TAB + (long)b * S * D;

    // t1 = gelu(rel @ W1 + b1)        [128,64]
    gemm128_wmma(relb, 4, W1, b1, t1b, HID, 1, nullptr);
    __threadfence(); __syncthreads();
    // h = t1 @ W2 + b2                [128,256]
    gemm128_wmma(t1b, HID, W2, b2, hb, D, 0, nullptr);
    __threadfence(); __syncthreads();
    // a1 = h @ W3 + b3                [128,256]
    gemm128_wmma(hb, D, W3, b3, a1b, D, 0, nullptr);
    __threadfence(); __syncthreads();

    // g = gelu(layernorm(a1))
    if (threadIdx.x < S) {
        const float* x = a1b + (long)threadIdx.x * D;
        float mu = 0.0f;
        for (int j = 0; j < D; j++) mu += x[j];
        mu *= (1.0f / D);
        float var = 0.0f;
        for (int j = 0; j < D; j++) { float d = x[j] - mu; var += d * d; }
        var *= (1.0f / D);
        float rs = rsqrtf(var + LN_EPS);
        float* y = gb + (long)threadIdx.x * D;
        for (int j = 0; j < D; j++)
            y[j] = gelu_exact((x[j] - mu) * rs * ln_g[j] + ln_b[j]);
    }
    __threadfence(); __syncthreads();

    // table = (g @ W4 + b4) * rowmask  [128,256]
    gemm128_wmma(gb, D, W4, b4, tab, D, 0, mskb);
}

// ------------------------- K4: streaming gather ----------------------------
// out[b,n,:] = table[b, labels[b,n], :].  327 MB write -> NT stores.
__global__ __launch_bounds__(256) void k4_gather(
    const long long* __restrict__ labels,
    const float* __restrict__ ws,
    float* __restrict__ out) {
    const f4* __restrict__ tab4 = (const f4*)(ws + WS_TAB);
    f4* __restrict__ out4 = (f4*)out;
    const int V = D / 4;  // 64 float4 per row
    const long total = (long)B * N * V;
    const long stride = (long)gridDim.x * blockDim.x;
    for (long i = (long)blockIdx.x * blockDim.x + threadIdx.x; i < total; i += stride) {
        const int c = (int)(i & (V - 1));
        const long nb = i >> 6;               // = b*N + n
        const int n = (int)(nb % N);
        const int b = (int)(nb / N);
        const int lab = (int)labels[(long)b * N + n];
        f4 v = tab4[((long)b * S + lab) * V + c];
        __builtin_nontemporal_store(v, &out4[i]);
    }
}

// ---------------------------------------------------------------------------
extern "C" void kernel_launch(void* const* d_in, const int* in_sizes, int n_in,
                              void* d_out, int out_size, void* d_ws, size_t ws_size,
                              hipStream_t stream) {
    const float*     coords = (const float*)d_in[0];
    // d_in[1] = features: unused by the reference computation
    const long long* labels = (const long long*)d_in[2];  // int64 per reference
    // d_in[3] = num_superpoints scalar (S=128, compile-time)
    const float* W1 = (const float*)d_in[4];
    const float* b1 = (const float*)d_in[5];
    const float* W2 = (const float*)d_in[6];
    const float* b2 = (const float*)d_in[7];
    const float* W3 = (const float*)d_in[8];
    const float* b3 = (const float*)d_in[9];
    const float* lg = (const float*)d_in[10];
    const float* lb = (const float*)d_in[11];
    const float* W4 = (const float*)d_in[12];
    const float* b4 = (const float*)d_in[13];
    float* out = (float*)d_out;
    float* ws  = (float*)d_ws;

    k0_zero_acc<<<(B * S * 4 + 255) / 256, 256, 0, stream>>>(ws);
    k1_segment_sum<<<B * CHUNKS, 256, 0, stream>>>(coords, labels, ws);
    k2_rel<<<B, S, 0, stream>>>(ws);
    k3_mlp<<<B, 256, 0, stream>>>(ws, W1, b1, W2, b2, W3, b3, lg, lb, W4, b4);
    k4_gather<<<4096, 256, 0, stream>>>(labels, ws, out);
}